// MPSModel_17927193493977
// MI455X (gfx1250) — compile-verified
//
#include <hip/hip_runtime.h>
#include <hip/hip_bf16.h>

// MPS / tensor-train scan on MI455X (gfx1250, wave32).
//
// Per feature j:  V <- diag(cos_j) * (V @ C0_j) + diag(sin_j) * (V @ C1_j)
//
// Each wave owns a 16-row batch tile. State lives permanently in WMMA
// *A-layout* (lane l holds batch row M = l%16, bond K = 4*kk + v + 2*(l/16)),
// so the per-sample diagonal scaling uses each lane's OWN cos/sin — zero
// cross-lane shuffles in the loop. Per step:
//   preload next B operands  (17 global loads issued one step ahead: latency
//                             hidden behind this step's WMMA+LDS+combine)
//   U0 = V@C0, U1 = V@C1     (2 x 4 chained V_WMMA_F32_16X16X4_F32)
//   D-layout -> LDS          (4 x ds_store_b128, conflict-free stride 20)
//   A-layout <- LDS, scaled  (merged ds_load_2addr_b32 + lane-own cos/sin fma)
//
// All 16 B loads share ONE per-lane address vector (lane-variant byte offset
// is lm*4 + half*256; the (kk,v) part is a constant instruction offset).

typedef float v2f __attribute__((ext_vector_type(2)));
typedef float v4f __attribute__((ext_vector_type(4)));
typedef float v8f __attribute__((ext_vector_type(8)));

#define NF      2000
#define NMID    (NF - 2)      // 1998 (even -> clean unroll-by-2)
#define WAVES   8
#define TSTRIDE 20            // floats per column; gcd(20,64)=4 -> conflict-free
#define TILE    (16 * TSTRIDE)

// One scan step: consume current B operands (bc0/bc1), preload next (bn0/bn1)
// from nbp, update A-layout state `a` in place.
__device__ __forceinline__ void mps_step(
    v2f a[4],
    const v2f bc0[4], const v2f bc1[4],
    v2f bn0[4], v2f bn1[4], const float* nbp,
    float th, float* t0, float* t1, int lm, int half)
{
    // Issue next-iteration global loads FIRST so they overlap this whole step.
    #pragma unroll
    for (int kk = 0; kk < 4; ++kk)
        #pragma unroll
        for (int v = 0; v < 2; ++v) {
            bn0[kk][v] = nbp[(4 * kk + v) * 32];
            bn1[kk][v] = nbp[(4 * kk + v) * 32 + 16];
        }
    __builtin_prefetch(nbp + 512, 0, 0);   // two blocks ahead into cache

    float cj = __cosf(th), sj = __sinf(th);

    // Two independent 16x16x16 f32 GEMM chains (4 x K=4 WMMA each)
    v8f u0 = {};
    v8f u1 = {};
    #pragma unroll
    for (int kk = 0; kk < 4; ++kk) {
        u0 = __builtin_amdgcn_wmma_f32_16x16x4_f32(
                 false, a[kk], false, bc0[kk], (short)0, u0, false, false);
        u1 = __builtin_amdgcn_wmma_f32_16x16x4_f32(
                 false, a[kk], false, bc1[kk], (short)0, u1, false, false);
    }

    // D-layout -> LDS: lane holds column e = lm, rows M = 8*half + r
    v4f lo0, hi0, lo1, hi1;
    #pragma unroll
    for (int r = 0; r < 4; ++r) {
        lo0[r] = u0[r]; hi0[r] = u0[r + 4];
        lo1[r] = u1[r]; hi1[r] = u1[r + 4];
    }
    v4f* d0 = (v4f*)(t0 + lm * TSTRIDE + 8 * half);  // bytes 80*lm+32*half: 16B ok
    v4f* d1 = (v4f*)(t1 + lm * TSTRIDE + 8 * half);
    d0[0] = lo0; d0[1] = hi0;
    d1[0] = lo1; d1[1] = hi1;

    // A-layout readback + diagonal scaling with the lane's OWN cos/sin
    #pragma unroll
    for (int kk = 0; kk < 4; ++kk)
        #pragma unroll
        for (int v = 0; v < 2; ++v) {
            const int K = 4 * kk + v + 2 * half;
            a[kk][v] = cj * t0[K * TSTRIDE + lm] + sj * t1[K * TSTRIDE + lm];
        }
}

__global__ __launch_bounds__(256, 1) void mps_tt_scan(
    const float* __restrict__ x,           // [BATCH, NF]
    const float* __restrict__ core_first,  // [1,2,16]   (32 floats)
    const float* __restrict__ cores_mid,   // [NMID,16,2,16]
    const float* __restrict__ core_last,   // [16,2,1]   (32 floats)
    float* __restrict__ out,               // [BATCH]
    int batch)
{
    __shared__ __align__(16) float lds[WAVES * 2 * TILE];

    const int lane = threadIdx.x & 31;
    const int wave = threadIdx.x >> 5;
    const int half = lane >> 4;            // which 16-lane half of the wave
    const int lm   = lane & 15;            // batch row within tile (A-layout M)
    const int row0 = (blockIdx.x * WAVES + wave) * 16;
    if (row0 + 16 > batch) return;         // whole-wave uniform: EXEC stays full

    float* t0 = lds + wave * 2 * TILE;     // U0 tile  [bond e][batch M]
    float* t1 = t0 + TILE;                 // U1 tile

    const float HPI = 1.5707963267948966f;
    const float* xrow = x + (size_t)(row0 + lm) * NF;   // this lane's batch row

    // ---------- feature 0: build V directly in A-layout (lane-own cos/sin)
    float th0 = xrow[0] * HPI;
    float c = __cosf(th0), s = __sinf(th0);
    v2f a[4];
    #pragma unroll
    for (int kk = 0; kk < 4; ++kk)
        #pragma unroll
        for (int v = 0; v < 2; ++v) {
            const int K = 4 * kk + v + 2 * half;
            a[kk][v] = c * core_first[K] + s * core_first[16 + K];
        }

    // Per-lane base pointer for B loads (block of feature 1).
    const float* bptr = cores_mid + lm + 64 * half;

    // Prime the pipeline: B operands and x for feature 1.
    v2f bA0[4], bA1[4], bB0[4], bB1[4];
    #pragma unroll
    for (int kk = 0; kk < 4; ++kk)
        #pragma unroll
        for (int v = 0; v < 2; ++v) {
            bA0[kk][v] = bptr[(4 * kk + v) * 32];
            bA1[kk][v] = bptr[(4 * kk + v) * 32 + 16];
        }
    float xv = xrow[1];

    // ---------- middle features: software-pipelined scan, unrolled by 2
    for (int j = 1; j <= NMID; j += 2) {
        float xv2 = xrow[j + 1];                       // j+1 <= NMID always
        mps_step(a, bA0, bA1, bB0, bB1, bptr + 512,    // preload block j+1
                 xv * HPI, t0, t1, lm, half);

        const bool more = (j + 2 <= NMID);
        const float* p3 = bptr + (more ? 1024 : 512);  // clamp final preload
        float xv3 = xrow[more ? (j + 2) : (j + 1)];
        mps_step(a, bB0, bB1, bA0, bA1, p3,            // preload block j+2
                 xv2 * HPI, t0, t1, lm, half);

        xv = xv3;
        bptr += 1024;
    }

    // ---------- last feature: out[b] = sum_d V[b,d] * (cos_b*L0[d] + sin_b*L1[d])
    // A-layout: this lane holds 8 of the 16 d-values of its own row; the other
    // 8 live in the partner lane (lane ^ 16). One shfl_xor(16) completes the dot.
    float thl = xrow[NF - 1] * HPI;
    float cl = __cosf(thl), sl = __sinf(thl);
    float t = 0.0f;
    #pragma unroll
    for (int kk = 0; kk < 4; ++kk)
        #pragma unroll
        for (int v = 0; v < 2; ++v) {
            const int K = 4 * kk + v + 2 * half;
            float w = cl * core_last[2 * K] + sl * core_last[2 * K + 1];
            t += a[kk][v] * w;
        }
    t += __shfl_xor(t, 16, 32);
    if (half == 0) out[row0 + lm] = t;
}

extern "C" void kernel_launch(void* const* d_in, const int* in_sizes, int n_in,
                              void* d_out, int out_size, void* d_ws, size_t ws_size,
                              hipStream_t stream) {
    const float* x  = (const float*)d_in[0];   // [BATCH, 2000] f32
    const float* cf = (const float*)d_in[1];   // [1,2,16]      f32
    const float* cm = (const float*)d_in[2];   // [1998,16,2,16]f32
    const float* cl = (const float*)d_in[3];   // [16,2,1]      f32
    float* out = (float*)d_out;

    const int batch = in_sizes[0] / NF;        // 4096
    const int rows_per_block = 16 * WAVES;     // 128
    const int grid = (batch + rows_per_block - 1) / rows_per_block;  // 32

    mps_tt_scan<<<grid, 256, 0, stream>>>(x, cf, cm, cl, out, batch);
}